// firstCapsuleLayer_10797547782885
// MI455X (gfx1250) — compile-verified
//
#include <hip/hip_runtime.h>

typedef float v2f __attribute__((ext_vector_type(2)));
typedef float v8f __attribute__((ext_vector_type(8)));

#define MROWS    (128 * 512)   // B*N output rows
#define KDIM     256           // F
#define CDIM     256           // H*Dh output channels
#define RT       32            // rows per tile
#define PITCH    260           // padded LDS pitch (even -> b64 aligned, breaks bank stride)
#define NTHREADS 512           // 16 wave32 waves
#define NBLOCKS  512           // persistent blocks; 2048 tiles total -> 4 each

#define USE_ASYNC_LDS_COPY 1   // gfx1250 GLOBAL_LOAD_ASYNC_TO_LDS_B128 path

// LDS float footprint
#define LDS_FLOATS (CDIM*PITCH + RT*PITCH + KDIM + KDIM + CDIM + RT + RT)

// ds_swizzle_b32 immediate: group-of-32 mode, and_mask=0x1f, or_mask=0, xor_mask=m
#define SWZ_XOR(x, m) __builtin_amdgcn_ds_swizzle((x), (0x1F | ((m) << 10)))

__device__ __forceinline__ float swz_xor_add(float s, int m_times_1024_plus_1f_unused) {
    return s; // unused helper guard (kept trivial)
}

__global__ __launch_bounds__(NTHREADS)
void firstCapsuleLayer_kernel(const float* __restrict__ x,
                              const unsigned char* __restrict__ mask,  // jax bool -> 1 byte
                              const float* __restrict__ gamma,
                              const float* __restrict__ beta,
                              const float* __restrict__ rmean,
                              const float* __restrict__ rvar,
                              const float* __restrict__ W,     // [H, F, Dh] = [8,256,32]
                              const float* __restrict__ bvec,  // [H, Dh] flat = [256]
                              float* __restrict__ out)         // [MROWS, 256]
{
    extern __shared__ float lds[];
    float* Wt    = lds;                       // [CDIM][PITCH]  BN-scaled, transposed weights
    float* Xt    = Wt + CDIM * PITCH;         // [RT][PITCH]    raw x tile
    float* bnS   = Xt + RT * PITCH;           // [KDIM] scale
    float* bnT   = bnS + KDIM;                // [KDIM] shift
    float* cbias = bnT + KDIM;                // [CDIM] bias + folded BN shift term
    float* sqs   = cbias + CDIM;              // [RT] per-row sum of squares
    float* mrow  = sqs + RT;                  // [RT] row mask as float

    const int tid  = threadIdx.x;
    const int lane = tid & 31;
    const int wave = tid >> 5;

    // ---- Phase 0: BN folding scalars ----
    for (int f = tid; f < KDIM; f += NTHREADS) {
        float s = gamma[f] * rsqrtf(rvar[f] + 1e-5f);
        bnS[f] = s;
        bnT[f] = beta[f] - rmean[f] * s;
    }
    for (int c = tid; c < CDIM; c += NTHREADS) cbias[c] = bvec[c];
    __syncthreads();

    // ---- Phase 1: Wt[c][k] = W[h][k][d] * bnS[k];  cbias[c] += sum_k bnT[k]*W[h][k][d]
    {
        const int c = tid & (CDIM - 1);       // invariant across the strided loop
        const int h = c >> 5, d = c & 31;
        float acc = 0.f;
        for (int i = tid; i < CDIM * KDIM; i += NTHREADS) {
            int k = i >> 8;                   // KDIM == 256
            float w = W[(h * KDIM + k) * 32 + d];
            Wt[c * PITCH + k] = w * bnS[k];
            acc += w * bnT[k];
        }
        atomicAdd(&cbias[c], acc);            // two threads per c
    }
    __syncthreads();

    // ---- wave -> output slab mapping ----
    const int rowhalf = wave & 1;             // which 16-row half of the 32-row tile
    const int cstrip  = wave >> 1;            // 8 strips of 32 columns
    const int c0      = cstrip * 32;
    // ISA f32 A 16x4 layout: lanes 0-15 hold K=0,1; lanes 16-31 hold K=2,3; M = lane&15
    const int koff = (lane >> 4) << 1;
    const float* Abase  = &Xt[((rowhalf << 4) + (lane & 15)) * PITCH + koff];
    const float* B0base = &Wt[(c0 + (lane & 15)) * PITCH + koff];
    const float* B1base = B0base + 16 * PITCH;
    const float cb0 = cbias[c0 + (lane & 15)];
    const float cb1 = cbias[c0 + 16 + (lane & 15)];
    // C/D layout: lane l, VGPR v -> tile row = rowhalf*16 + (l>>4)*8 + v, col = c0 + (l&15)
    const int rbase = (rowhalf << 4) + ((lane >> 4) << 3);

    for (int tile = blockIdx.x; tile < MROWS / RT; tile += gridDim.x) {
        const int row0 = tile * RT;
        __syncthreads();  // previous iteration finished reading Xt/sqs/mrow

#if USE_ASYNC_LDS_COPY
        // raw x tile -> LDS via CDNA5 async copy (ASYNCcnt path, no VGPR round trip)
        for (int i = tid * 4; i < RT * KDIM; i += NTHREADS * 4) {
            int r = i >> 8, f = i & 255;
            unsigned ldsOff = (unsigned)(unsigned long long)(&Xt[r * PITCH + f]);
            const float* g = &x[(size_t)(row0 + r) * KDIM + f];
            asm volatile("global_load_async_to_lds_b128 %0, %1, off"
                         :: "v"(ldsOff), "v"(g)
                         : "memory");
        }
#else
        for (int i = tid * 4; i < RT * KDIM; i += NTHREADS * 4) {
            int r = i >> 8, f = i & 255;
            *(float4*)&Xt[r * PITCH + f] =
                *(const float4*)&x[(size_t)(row0 + r) * KDIM + f];
        }
#endif
        if (tid < RT) {
            sqs[tid]  = 0.f;
            mrow[tid] = mask[row0 + tid] ? 1.f : 0.f;
        }
#if USE_ASYNC_LDS_COPY
        asm volatile("s_wait_asynccnt 0x0" ::: "memory");
#endif
        __syncthreads();

        v8f acc0, acc1;
        #pragma unroll
        for (int v = 0; v < 8; ++v) { acc0[v] = cb0; acc1[v] = cb1; }

        #pragma unroll 8
        for (int k0 = 0; k0 < KDIM; k0 += 4) {
            v2f a  = *(const v2f*)(Abase  + k0);   // ds_load_b64
            v2f b0 = *(const v2f*)(B0base + k0);
            v2f b1 = *(const v2f*)(B1base + k0);
            acc0 = __builtin_amdgcn_wmma_f32_16x16x4_f32(
                       false, a, false, b0, (short)0, acc0, false, false);
            acc1 = __builtin_amdgcn_wmma_f32_16x16x4_f32(
                       false, a, false, b1, (short)0, acc1, false, false);
        }

        // relu + mask + per-row sum of squares (ds_swizzle XOR butterfly + LDS float atomic)
        #pragma unroll
        for (int v = 0; v < 8; ++v) {
            const int rloc = rbase + v;
            float m  = mrow[rloc];
            float t0 = fmaxf(acc0[v], 0.f) * m;
            float t1 = fmaxf(acc1[v], 0.f) * m;
            acc0[v] = t0; acc1[v] = t1;
            float s = t0 * t0 + t1 * t1;
            s += __int_as_float(SWZ_XOR(__float_as_int(s), 1));
            s += __int_as_float(SWZ_XOR(__float_as_int(s), 2));
            s += __int_as_float(SWZ_XOR(__float_as_int(s), 4));
            s += __int_as_float(SWZ_XOR(__float_as_int(s), 8));
            if ((lane & 15) == 0) atomicAdd(&sqs[rloc], s);  // ds_add_f32
        }
        __syncthreads();

        // squash scale + store
        #pragma unroll
        for (int v = 0; v < 8; ++v) {
            const int rloc = rbase + v;
            float sq  = sqs[rloc];
            float fac = sq / ((1.f + sq) * sqrtf(sq + 1e-7f));
            size_t o = (size_t)(row0 + rloc) * CDIM + c0 + (lane & 15);
            out[o]      = acc0[v] * fac;
            out[o + 16] = acc1[v] * fac;
        }
    }
}

extern "C" void kernel_launch(void* const* d_in, const int* in_sizes, int n_in,
                              void* d_out, int out_size, void* d_ws, size_t ws_size,
                              hipStream_t stream) {
    // setup_inputs order: x, adj, mask, batch, gamma, beta, running_mean, running_var, W, b
    const float*         x     = (const float*)d_in[0];
    // d_in[1] adj   : unused by the forward
    const unsigned char* mask  = (const unsigned char*)d_in[2];  // jax bool -> byte
    // d_in[3] batch : unused by the forward
    const float*         gamma = (const float*)d_in[4];
    const float*         beta  = (const float*)d_in[5];
    const float*         rmean = (const float*)d_in[6];
    const float*         rvar  = (const float*)d_in[7];
    const float*         W     = (const float*)d_in[8];
    const float*         bvec  = (const float*)d_in[9];
    float*               out   = (float*)d_out;

    const size_t lds_bytes = (size_t)LDS_FLOATS * sizeof(float);  // ~296 KB, < 320 KB/WGP
    firstCapsuleLayer_kernel<<<NBLOCKS, NTHREADS, lds_bytes, stream>>>(
        x, mask, gamma, beta, rmean, rvar, W, bvec, out);
}